// VQModel_68719477172
// MI455X (gfx1250) — compile-verified
//
#include <hip/hip_runtime.h>
#include <hip/hip_bf16.h>

typedef float v2f __attribute__((ext_vector_type(2)));
typedef float v8f __attribute__((ext_vector_type(8)));

#define N_ROWS   8192
#define K_CODES  16384
#define DIM      8
#define TWO_T    200.0f      // 2 / ENT_TEMP
#define VQ_EPS   1e-5f
#define NEG_INF  (-3.0e38f)

// workspace layout (float offsets)
#define WS_ZF     0          // [8192][8] packed z rows (NHWC-flat)
#define WS_CN100  65536      // [16384]  100 * ||e_k||^2
#define WS_ROWT   81920      // [8192]   t_n = max_k (200*dot - 100*||e||^2)
#define WS_INVL   90112      // [8192]   1 / l_n
#define WS_AVGP   98304      // [16384]  avg_probs

// output layout (floats): z_q[65536], loss[1], idx[8192]
#define OUT_ZQ    0
#define OUT_LOSS  65536
#define OUT_IDX   65537

__device__ __forceinline__ v8f wmma_f32_k4(v2f a, v2f b, v8f c) {
  return __builtin_amdgcn_wmma_f32_16x16x4_f32(
      /*neg_a=*/false, a, /*neg_b=*/false, b,
      /*c_mod=*/(short)0, c, /*reuse_a=*/false, /*reuse_b=*/false);
}

// ---------------- kernel 0: pack z + scaled codebook norms + loss init ----------------
__global__ __launch_bounds__(256) void vq_prep(const float* __restrict__ z,
                                               const float* __restrict__ cb,
                                               float* __restrict__ ws,
                                               float* __restrict__ out) {
  int tid = blockIdx.x * blockDim.x + threadIdx.x;
  if (tid < N_ROWS) {
    // zf[n][c] = z[b][c][h][w], n = b*256 + h*16 + w
    int b = tid >> 8, hw = tid & 255;
    const float* zp = z + b * (DIM * 256) + hw;
#pragma unroll
    for (int c = 0; c < DIM; ++c) ws[WS_ZF + tid * DIM + c] = zp[c * 256];
  }
  if (tid < K_CODES) {
    float nrm = 0.f;
#pragma unroll
    for (int c = 0; c < DIM; ++c) { float v = cb[tid * DIM + c]; nrm += v * v; }
    ws[WS_CN100 + tid] = 100.0f * nrm;
  }
  if (tid == 0) out[OUT_LOSS] = 0.f;
}

// ---------------- kernel 1: argmax sweep (no transcendentals) ----------------
// score(n,k) = 200*dot(z_n, e_k) - 100*||e_k||^2 ; argmax score == argmin dist,
// and max score == t_n (the softmax shift with ||z||^2 cancelled).
__global__ __launch_bounds__(256) void vq_pass1(const float* __restrict__ cb,
                                                float* __restrict__ ws,
                                                float* __restrict__ out) {
  __shared__ float sbs[8][16];
  __shared__ int   sbi[8][16];

  const int lane = threadIdx.x & 31;
  const int wave = threadIdx.x >> 5;
  const int lmod = lane & 15;
  const int hi   = lane >> 4;
  const int kb   = hi * 2;
  const int row0 = blockIdx.x * 16;

  const float* zrow = ws + WS_ZF + (row0 + lmod) * DIM;
  v2f a0, a1;
  a0[0] = zrow[kb];     a0[1] = zrow[kb + 1];
  a1[0] = zrow[kb + 4]; a1[1] = zrow[kb + 5];

  float bs[8]; int bi[8];
#pragma unroll
  for (int v = 0; v < 8; ++v) { bs[v] = NEG_INF; bi[v] = 0; }

  const int col_base = wave * 2048;
  for (int t = 0; t < 128; ++t) {
    const int col = col_base + t * 16 + lmod;
    const float* crow = cb + col * DIM;
    v2f b0, b1;
    b0[0] = crow[kb];     b0[1] = crow[kb + 1];
    b1[0] = crow[kb + 4]; b1[1] = crow[kb + 5];
    const float negcn = -ws[WS_CN100 + col];

    v8f c = {0.f, 0.f, 0.f, 0.f, 0.f, 0.f, 0.f, 0.f};
    c = wmma_f32_k4(a0, b0, c);
    c = wmma_f32_k4(a1, b1, c);

#pragma unroll
    for (int v = 0; v < 8; ++v) {
      float score = __builtin_fmaf(c[v], TWO_T, negcn);
      if (score > bs[v]) { bs[v] = score; bi[v] = col; }
    }
  }

  // reduce across the 16 lanes of each row half (ties -> smaller index)
#pragma unroll
  for (int mask = 1; mask <= 8; mask <<= 1) {
#pragma unroll
    for (int v = 0; v < 8; ++v) {
      float obs = __shfl_xor(bs[v], mask, 32);
      int   obi = __shfl_xor(bi[v], mask, 32);
      if (obs > bs[v] || (obs == bs[v] && obi < bi[v])) { bs[v] = obs; bi[v] = obi; }
    }
  }
  if (lmod == 0) {
#pragma unroll
    for (int v = 0; v < 8; ++v) {
      int r = v + 8 * hi;
      sbs[wave][r] = bs[v]; sbi[wave][r] = bi[v];
    }
  }
  __syncthreads();

  if (threadIdx.x < 16) {
    const int r = threadIdx.x;
    float BS = sbs[0][r]; int BI = sbi[0][r];
    for (int w = 1; w < 8; ++w) {
      float obs = sbs[w][r]; int obi = sbi[w][r];
      if (obs > BS || (obs == BS && obi < BI)) { BS = obs; BI = obi; }
    }
    const int n = row0 + r;
    ws[WS_ROWT + n] = BS;                 // t_n for passes 2a/2b
    // vq + commitment = (1 + beta) * mean((zq - z)^2)
    const float* e  = cb + BI * DIM;
    const float* zr = ws + WS_ZF + n * DIM;
    float sq = 0.f;
#pragma unroll
    for (int c = 0; c < DIM; ++c) { float d = e[c] - zr[c]; sq += d * d; }
    float contrib = 1.25f * sq / (float)(N_ROWS * DIM);
    // straight-through z_q back to NCHW + index output
    int b = n >> 8, hw = n & 255;
    float* oz = out + OUT_ZQ + b * 2048 + hw;
#pragma unroll
    for (int c = 0; c < DIM; ++c) oz[c * 256] = e[c];
    out[OUT_IDX + n] = (float)BI;
#pragma unroll
    for (int mask = 1; mask <= 8; mask <<= 1) contrib += __shfl_xor(contrib, mask, 32);
    if (r == 0) atomicAdd(out + OUT_LOSS, contrib);
  }
}

// ---------------- kernel 2a: row softmax stats (1 exp / element) ----------------
__global__ __launch_bounds__(256) void vq_pass2a(const float* __restrict__ cb,
                                                 float* __restrict__ ws,
                                                 float* __restrict__ out) {
  __shared__ float sl[8][16], sw[8][16];

  const int lane = threadIdx.x & 31;
  const int wave = threadIdx.x >> 5;
  const int lmod = lane & 15;
  const int hi   = lane >> 4;
  const int kb   = hi * 2;
  const int row0 = blockIdx.x * 16;

  const float* zrow = ws + WS_ZF + (row0 + lmod) * DIM;
  v2f a0, a1;
  a0[0] = zrow[kb];     a0[1] = zrow[kb + 1];
  a1[0] = zrow[kb + 4]; a1[1] = zrow[kb + 5];

  float tv[8];
#pragma unroll
  for (int v = 0; v < 8; ++v) tv[v] = ws[WS_ROWT + row0 + v + 8 * hi];

  float l_[8], wu[8];
#pragma unroll
  for (int v = 0; v < 8; ++v) { l_[v] = 0.f; wu[v] = 0.f; }

  const int col_base = wave * 2048;
  for (int t = 0; t < 128; ++t) {
    const int col = col_base + t * 16 + lmod;
    const float* crow = cb + col * DIM;
    v2f b0, b1;
    b0[0] = crow[kb];     b0[1] = crow[kb + 1];
    b1[0] = crow[kb + 4]; b1[1] = crow[kb + 5];
    const float negcn = -ws[WS_CN100 + col];

    v8f c = {0.f, 0.f, 0.f, 0.f, 0.f, 0.f, 0.f, 0.f};
    c = wmma_f32_k4(a0, b0, c);
    c = wmma_f32_k4(a1, b1, c);

#pragma unroll
    for (int v = 0; v < 8; ++v) {
      float u = __builtin_fmaf(c[v], TWO_T, negcn) - tv[v];  // = s - m_n  (<= 0)
      float e = __expf(u);
      l_[v] += e;
      wu[v] = __builtin_fmaf(e, u, wu[v]);
    }
  }

  // plain-sum reductions within each row half
#pragma unroll
  for (int mask = 1; mask <= 8; mask <<= 1) {
#pragma unroll
    for (int v = 0; v < 8; ++v) {
      l_[v] += __shfl_xor(l_[v], mask, 32);
      wu[v] += __shfl_xor(wu[v], mask, 32);
    }
  }
  if (lmod == 0) {
#pragma unroll
    for (int v = 0; v < 8; ++v) {
      int r = v + 8 * hi;
      sl[wave][r] = l_[v]; sw[wave][r] = wu[v];
    }
  }
  __syncthreads();

  if (threadIdx.x < 16) {
    const int r = threadIdx.x;
    float L = 0.f, WU = 0.f;
    for (int w = 0; w < 8; ++w) { L += sl[w][r]; WU += sw[w][r]; }
    const int n = row0 + r;
    ws[WS_INVL + n] = 1.f / L;
    // sum_k p*log p = WU/L - log L ; sample_entropy piece of the loss
    float contrib = -(0.1f / (float)N_ROWS) * (WU / L - __logf(L));
#pragma unroll
    for (int mask = 1; mask <= 8; mask <<= 1) contrib += __shfl_xor(contrib, mask, 32);
    if (r == 0) atomicAdd(out + OUT_LOSS, contrib);
  }
}

// ---------------- kernel 2b: avg_probs (column-stationary, atomic-free) ----------------
__global__ __launch_bounds__(256) void vq_pass2b(const float* __restrict__ cb,
                                                 float* __restrict__ ws) {
  const int lane = threadIdx.x & 31;
  const int wave = threadIdx.x >> 5;
  const int lmod = lane & 15;
  const int hi   = lane >> 4;
  const int kb   = hi * 2;
  const int col  = blockIdx.x * 128 + wave * 16 + lmod;

  const float* crow = cb + col * DIM;
  v2f b0, b1;
  b0[0] = crow[kb];     b0[1] = crow[kb + 1];
  b1[0] = crow[kb + 4]; b1[1] = crow[kb + 5];
  const float negcn = -ws[WS_CN100 + col];

  float acc = 0.f;
  for (int rt = 0; rt < N_ROWS / 16; ++rt) {
    const int row0 = rt * 16;
    const float* zrow = ws + WS_ZF + (row0 + lmod) * DIM;
    v2f a0, a1;
    a0[0] = zrow[kb];     a0[1] = zrow[kb + 1];
    a1[0] = zrow[kb + 4]; a1[1] = zrow[kb + 5];

    v8f c = {0.f, 0.f, 0.f, 0.f, 0.f, 0.f, 0.f, 0.f};
    c = wmma_f32_k4(a0, b0, c);
    c = wmma_f32_k4(a1, b1, c);

#pragma unroll
    for (int v = 0; v < 8; ++v) {
      const int n = row0 + v + 8 * hi;
      float t  = ws[WS_ROWT + n];
      float il = ws[WS_INVL + n];
      acc += __expf(__builtin_fmaf(c[v], TWO_T, negcn) - t) * il;
    }
  }
  acc += __shfl_xor(acc, 16, 32);
  if (hi == 0) ws[WS_AVGP + col] = acc * (1.f / (float)N_ROWS);
}

// ---------------- kernel 3: avg-entropy term ----------------
__global__ __launch_bounds__(256) void vq_pass3(const float* __restrict__ ws,
                                                float* __restrict__ out) {
  __shared__ float red[256];
  float acc = 0.f;
  for (int k = threadIdx.x; k < K_CODES; k += 256) {
    float q = ws[WS_AVGP + k];
    acc += q * __logf(q + VQ_EPS);
  }
  red[threadIdx.x] = acc;
  __syncthreads();
  for (int s = 128; s > 0; s >>= 1) {
    if (threadIdx.x < s) red[threadIdx.x] += red[threadIdx.x + s];
    __syncthreads();
  }
  // ent_loss += -0.1 * avg_entropy = +0.1 * sum q*log(q+eps)
  if (threadIdx.x == 0) atomicAdd(out + OUT_LOSS, 0.1f * red[0]);
}

extern "C" void kernel_launch(void* const* d_in, const int* in_sizes, int n_in,
                              void* d_out, int out_size, void* d_ws, size_t ws_size,
                              hipStream_t stream) {
  const float* z  = (const float*)d_in[0];   // [32,8,16,16]
  const float* cb = (const float*)d_in[1];   // [16384,8]
  float* out = (float*)d_out;
  float* ws  = (float*)d_ws;                 // needs ~448 KB

  vq_prep  <<<64,  256, 0, stream>>>(z, cb, ws, out);
  vq_pass1 <<<N_ROWS / 16, 256, 0, stream>>>(cb, ws, out);
  vq_pass2a<<<N_ROWS / 16, 256, 0, stream>>>(cb, ws, out);
  vq_pass2b<<<K_CODES / 128, 256, 0, stream>>>(cb, ws);
  vq_pass3 <<<1,   256, 0, stream>>>(ws, out);
}